// NuHatFlatPlateSolver_83769042141606
// MI455X (gfx1250) — compile-verified
//
#include <hip/hip_runtime.h>
#include <math.h>

#define NXS 160
#define NYS 80
#define NBATCH 64
#define NNEWT 45
#define BLK 96

typedef __attribute__((ext_vector_type(2))) float v2f;
typedef __attribute__((ext_vector_type(8))) float v8f;

// ---------------------------------------------------------------------------
// SA model helpers (constants folded: cv1^3=357.911, kappa^2=0.1681,
// cw1 = cb1/kappa^2 + (1+cb2)/sigma = 3.2390678, cw3^6 = 64, cb2/sigma=0.933)
// ---------------------------------------------------------------------------
__device__ __forceinline__ float fv1_of(float nu) {
    float nu3 = nu * nu * nu;
    return nu3 / (nu3 + 357.911f);
}

// Production P, dP/dnu, Destruction D, dD/dnu (hand-derived jvp of reference)
__device__ __forceinline__ void sa_terms(float nu, float absd, float ky2,
                                         float inv_ky2, float invy2,
                                         float& P, float& Pg, float& Dst, float& Dg) {
    float nu2 = nu * nu, nu3 = nu2 * nu;
    float q    = nu3 + 357.911f;
    float invq = 1.0f / q;
    float fv1  = nu3 * invq;
    float fv1p = 3.0f * nu2 * 357.911f * invq * invq;
    float den    = 1.0f + nu * fv1;
    float invden = 1.0f / den;
    float fv2  = 1.0f - nu * invden;
    float fv2p = -(1.0f - nu2 * fv1p) * invden * invden;
    float St   = absd + nu * fv2 * inv_ky2;            // s_tilde
    float Stk  = fv2 + nu * fv2p;                      // ky2 * dSt/dnu
    P  = 0.1355f * St * nu;
    Pg = 0.1355f * (St + nu * Stk * inv_ky2);
    float dR    = St * ky2 + 1e-12f;
    float invdR = 1.0f / dR;
    float rr    = nu * invdR;
    float r, rp;
    if (rr < 10.0f) { r = rr; rp = (dR - nu * Stk) * invdR * invdR; }
    else            { r = 10.0f; rp = 0.0f; }
    float r2 = r * r, r4 = r2 * r2, r5 = r4 * r, r6 = r4 * r2;
    float g  = r + 0.3f * (r6 - r);
    float gp = rp * (1.0f + 0.3f * (6.0f * r5 - 1.0f));
    float g2 = g * g, g6 = g2 * g2 * g2;
    float t   = g6 + 64.0f;
    float h   = powf(65.0f / t, 0.16666667f);          // ((1+cw3^6)/(g^6+cw3^6))^(1/6)
    float fw  = g * h;
    float fwp = gp * h * (64.0f / t);
    Dst = 3.2390678f * fw * nu2 * invy2;
    Dg  = 3.2390678f * invy2 * (fwp * nu2 + 2.0f * nu * fw);
}

// ---------------------------------------------------------------------------
// CDNA5 async copy: global -> LDS, tracked by ASYNCcnt (CDNA5 ISA)
// GVS mode: global address is the SGPR base plus the 32-bit VGPR offset; the async unit writes the LDS address in vdst.
// ---------------------------------------------------------------------------
__device__ __forceinline__ void async_ld(const float* gbase, float* ldsbase, int idx) {
    unsigned int lds  = (unsigned int)(unsigned long long)(ldsbase + idx); // low 32 bits of the pointer are the LDS offset
    unsigned int voff = (unsigned int)(idx * 4);
    asm volatile("global_load_async_to_lds_b32 %0, %1, %2"
                 :: "v"(lds), "v"(voff), "s"(gbase)
                 : "memory");
}
__device__ __forceinline__ void wait_async0() {
    asm volatile("s_wait_asynccnt 0" ::: "memory");
}

// ---------------------------------------------------------------------------
// Parallel cyclic reduction for one 80-row tridiagonal system.
// Double-buffered in LDS: one barrier per stride step, 7 steps (2^7 >= 80).
// Out-of-range neighbors act as identity rows (l=0,d=1,u=0,r=0).
// ---------------------------------------------------------------------------
__device__ __forceinline__ float pcr_solve(bool act, int i,
                                           float Ll, float Ld, float Lu, float rr,
                                           float sL[][NYS], float sD[][NYS],
                                           float sU[][NYS], float sR[][NYS]) {
    if (act) { sL[0][i] = Ll; sD[0][i] = Ld; sU[0][i] = Lu; sR[0][i] = rr; }
    __syncthreads();
    int c = 0;
#pragma unroll
    for (int s = 1; s < NYS; s <<= 1) {
        if (act) {
            float li = sL[c][i], di = sD[c][i], ui = sU[c][i], ri = sR[c][i];
            float lm = 0.f, dm = 1.f, um = 0.f, rm = 0.f;
            float lp = 0.f, dp = 1.f, up = 0.f, rp = 0.f;
            int im = i - s, ip = i + s;
            if (im >= 0)  { lm = sL[c][im]; dm = sD[c][im]; um = sU[c][im]; rm = sR[c][im]; }
            if (ip < NYS) { lp = sL[c][ip]; dp = sD[c][ip]; up = sU[c][ip]; rp = sR[c][ip]; }
            float a  = -li / dm;
            float bb = -ui / dp;
            int n = c ^ 1;
            sL[n][i] = a * lm;
            sU[n][i] = bb * up;
            sD[n][i] = di + a * um + bb * lp;
            sR[n][i] = ri + a * rm + bb * rp;
        }
        __syncthreads();
        c ^= 1;
    }
    return act ? (sR[c][i] / sD[c][i]) : 0.0f;
}

// ---------------------------------------------------------------------------
// One workgroup per batch element; lanes = y-points (80 active of 96).
// ---------------------------------------------------------------------------
__global__ __launch_bounds__(BLK)
void sa_march_kernel(const float* __restrict__ nuHat_in,
                     const float* __restrict__ u0_tpl,
                     const float* __restrict__ v0_tpl,
                     const float* __restrict__ dx_list,
                     const float* __restrict__ dy_vol,
                     const float* __restrict__ dy_dual,
                     const float* __restrict__ y_cell,
                     float* __restrict__ out) {
    __shared__ float s_dyv[NYS], s_dyd[NYS], s_y[NYS], s_dx[NXS];
    __shared__ float s_u[NYS], s_v[NYS], s_nh[NYS];
    __shared__ float s_u1[NYS], s_v1[NYS], s_n1[NYS], s_w[NYS];
    __shared__ float sL[2][NYS], sD[2][NYS], sU[2][NYS], sR[2][NYS];

    const int  tid = threadIdx.x;
    const int  b   = blockIdx.x;
    const bool act = (tid < NYS);
    const int  i   = tid;

    // Stage read-only grids into LDS via CDNA5 async-to-LDS path.
    if (act) {
        async_ld(dy_vol,  s_dyv, i);
        async_ld(dy_dual, s_dyd, i);
        async_ld(y_cell,  s_y,   i);
        async_ld(dx_list, s_dx,  i);
        async_ld(dx_list, s_dx,  i + NYS);
    }
    wait_async0();
    __syncthreads();

    const float nh_init = nuHat_in[b];

    float* outU  = out;
    float* outV  = out + (size_t)(NXS + 1) * NBATCH * NYS;
    float* outNH = outV + (size_t)(NXS + 1) * NBATCH * (NYS - 1);

    if (act) {
        float u0i = u0_tpl[i];
        float v0i = (i < NYS - 1) ? v0_tpl[i] : 0.0f;
        s_u[i]  = u0i;
        s_v[i]  = v0i;
        s_nh[i] = nh_init;
        size_t r0 = (size_t)b;                       // k = 0 slice
        outU [r0 * NYS + i] = u0i;
        outNH[r0 * NYS + i] = nh_init;
        if (i < NYS - 1) outV[r0 * (NYS - 1) + i] = v0i;
    }
    __syncthreads();

    for (int k = 0; k < NXS; ++k) {
        const float dx    = s_dx[k];
        const float dxinv = 1.0f / dx;

        // ---------------- momentum step ----------------
        float nh_i = 0, nh_m = 0, nh_p = 0, u0i = 0;
        float dyv = 1, dyd = 1, dyd_p = 1;
        float Ll = 0, Ld = 0, Lu = 0, rhs = 0;
        if (act) {
            nh_i = s_nh[i];
            nh_m = (i > 0)       ? s_nh[i - 1] : 0.0f;
            nh_p = (i < NYS - 1) ? s_nh[i + 1] : 0.0f;
            u0i  = s_u[i];
            float v_i = (i < NYS - 1) ? s_v[i] : 0.0f;
            float v_m = (i > 0)       ? s_v[i - 1] : 0.0f;
            dyv = s_dyv[i]; dyd = s_dyd[i];
            dyd_p = (i < NYS - 1) ? s_dyd[i + 1] : 1.0f;
            float ne_i = 1.0f + nh_i * fv1_of(nh_i);
            float ne_m = 1.0f + nh_m * fv1_of(nh_m);
            float ne_p = 1.0f + nh_p * fv1_of(nh_p);
            float nug_i = (i == 0) ? 1.0f : 0.5f * (ne_i + ne_m);
            float nug_p = 0.5f * (ne_p + ne_i);
            float cp = (i > 0)       ? fmaxf(v_m / dyd, 0.0f)    : 0.0f;
            float cm = (i < NYS - 1) ? fmaxf(-v_i / dyd_p, 0.0f) : 0.0f;
            float Dd = nug_i / (dyv * dyd);
            float Du = (i < NYS - 1) ? nug_p / (dyv * dyd_p) : 0.0f;
            Ld  = cp + cm + Dd + Du + u0i * dxinv;
            Ll  = (i > 0)       ? (-cp - Dd) : 0.0f;
            Lu  = (i < NYS - 1) ? (-cm - Du) : 0.0f;
            rhs = u0i * u0i * dxinv;
        }
        float u1i = pcr_solve(act, i, Ll, Ld, Lu, rhs, sL, sD, sU, sR);
        if (act) {
            s_u1[i] = u1i;
            s_w[i]  = (u1i - u0i) * dyv;             // integrand for v1 cumsum
        }
        __syncthreads();

        // ---- v1 = -cumsum(w)/dx as triangular matmul on the matrix pipe ----
        // P = W(16x16, rows>=5 zero) * T(upper-tri ones) via 4 chained
        // V_WMMA_F32_16X16X4_F32 (K=4 slices). Executed by ALL waves with full
        // EXEC (WMMA requires EXEC==all-ones); lanes 0..15 commit the result.
        {
            const int lane = tid & 31;
            const int m    = lane & 15;          // A row / B column / D column
            const int hi   = lane >> 4;          // K sub-pair selector
            v8f acc = {};
#pragma unroll
            for (int kk = 0; kk < 4; ++kk) {
                const int j0 = 16 * m + 4 * kk + 2 * hi;   // flat w index of A elem
                const int kr = 4 * kk + 2 * hi;            // B row of this pair
                v2f av, bv;
                av.x = (m < 5) ? s_w[j0]     : 0.0f;
                av.y = (m < 5) ? s_w[j0 + 1] : 0.0f;
                bv.x = (kr     <= m) ? 1.0f : 0.0f;        // T[kr][m]
                bv.y = (kr + 1 <= m) ? 1.0f : 0.0f;        // T[kr+1][m]
                acc = __builtin_amdgcn_wmma_f32_16x16x4_f32(
                        false, av, false, bv, (short)0, acc, false, false);
            }
            // Block offsets from row totals P[q][15] (lane 15 of each D row).
            float t0 = __shfl(acc[0], 15, 32);
            float t1 = __shfl(acc[1], 15, 32);
            float t2 = __shfl(acc[2], 15, 32);
            float t3 = __shfl(acc[3], 15, 32);
            float O1 = t0, O2 = O1 + t1, O3 = O2 + t2, O4 = O3 + t3;
            if (tid < 16) {                       // D: lane<16, VGPR v -> P[v][lane]
                s_v1[tid]      = -(acc[0])      * dxinv;
                s_v1[16 + tid] = -(acc[1] + O1) * dxinv;
                s_v1[32 + tid] = -(acc[2] + O2) * dxinv;
                s_v1[48 + tid] = -(acc[3] + O3) * dxinv;
                if (tid < 15) s_v1[64 + tid] = -(acc[4] + O4) * dxinv;
            }
        }
        __syncthreads();

        // ---------------- nuHat step: per-station precompute ----------------
        float Al = 0, Adg = 0, Au = 0, Ldb = 0, mdx = 0;
        float absd = 0, ky2 = 1, inv_ky2 = 1, invy2 = 1, is_turb = 0, aaft_c = 0;
        if (act) {
            float u1m  = (i > 0) ? s_u1[i - 1] : 0.0f;
            float y_i  = s_y[i];
            float dudy = (i == 0) ? (u1i / y_i) : ((u1i - u1m) / dyd);
            absd = fabsf(dudy);
            float gamma = absd * y_i / fabsf(u1i);
            float re_om = y_i * y_i * absd;
            float sg    = 1.0f / (1.0f + expf(10.0f - re_om * 0.01f));
            float a_aft = 0.04f * sg / (1.0f + gamma) * dudy;
            is_turb = fmaxf(1.0f - expf(-(nh_i - 1.0f) * 0.25f), 0.0f);
            aaft_c  = a_aft * nh_i * (1.0f - is_turb);
            float v1_i = (i < NYS - 1) ? s_v1[i] : 0.0f;
            float v1_m = (i > 0)       ? s_v1[i - 1] : 0.0f;
            float ve_i = v1_i - (nh_p - nh_i) / dyd_p * 0.933f;   // cb2/sigma
            float ve_m = v1_m - (nh_i - nh_m) / dyd   * 0.933f;
            float ne_i = (1.0f + nh_i) * 1.5f;                    // (1+nh)/sigma
            float ne_m = (1.0f + nh_m) * 1.5f;
            float ne_p = (1.0f + nh_p) * 1.5f;
            float nug_i = (i == 0) ? 1.0f : 0.5f * (ne_i + ne_m);
            float nug_p = 0.5f * (ne_p + ne_i);
            float cp = (i > 0)       ? fmaxf(ve_m / dyd, 0.0f)    : 0.0f;
            float cm = (i < NYS - 1) ? fmaxf(-ve_i / dyd_p, 0.0f) : 0.0f;
            float Dd = nug_i / (dyv * dyd);
            float Du = (i < NYS - 1) ? nug_p / (dyv * dyd_p) : 0.0f;
            Adg = cp + cm + Dd + Du;
            Al  = (i > 0)       ? (-cp - Dd) : 0.0f;
            Au  = (i < NYS - 1) ? (-cm - Du) : 0.0f;
            mdx = u1i * dxinv;
            Ldb = Adg + mdx;
            ky2     = 0.1681f * y_i * y_i;
            inv_ky2 = 1.0f / ky2;
            invy2   = 1.0f / (y_i * y_i);
        }

        // ---------------- Newton loop (45 damped iterations) ----------------
        float dn = 0.0f;
        for (int t = 0; t < NNEWT; ++t) {
            if (act) s_n1[i] = nh_i + dn;
            __syncthreads();
            float Ldl = 0, rr = 0;
            if (act) {
                float n1  = nh_i + dn;
                float n1m = (i > 0)       ? s_n1[i - 1] : 0.0f;
                float n1p = (i < NYS - 1) ? s_n1[i + 1] : 0.0f;
                float P, Pg, Dst, Dg;
                sa_terms(n1, absd, ky2, inv_ky2, invy2, P, Pg, Dst, Dg);
                float p  = P * is_turb + aaft_c;
                float pg = Pg * is_turb;
                rr  = mdx * dn + Al * n1m + Adg * n1 + Au * n1p - p + Dst;
                Ldl = Ldb + (Dg - pg);
            }
            float upd = pcr_solve(act, i, Al, Ldl, Au, rr, sL, sD, sU, sR);
            if (act) dn -= 0.9f * upd;
        }

        // ---------------- commit state + outputs ----------------
        __syncthreads();
        if (act) {
            float nh1 = nh_i + dn;
            s_u[i]  = u1i;
            s_nh[i] = nh1;
            if (i < NYS - 1) s_v[i] = s_v1[i];
            size_t row = (size_t)(k + 1) * NBATCH + (size_t)b;
            outU [row * NYS + i] = u1i;
            outNH[row * NYS + i] = nh1;
            if (i < NYS - 1) outV[row * (NYS - 1) + i] = s_v1[i];
        }
        __syncthreads();
    }
}

extern "C" void kernel_launch(void* const* d_in, const int* in_sizes, int n_in,
                              void* d_out, int out_size, void* d_ws, size_t ws_size,
                              hipStream_t stream) {
    (void)in_sizes; (void)n_in; (void)out_size; (void)d_ws; (void)ws_size;
    const float* nuHat = (const float*)d_in[0];
    const float* u0t   = (const float*)d_in[1];
    const float* v0t   = (const float*)d_in[2];
    const float* dxl   = (const float*)d_in[3];
    const float* dyv   = (const float*)d_in[4];
    const float* dyd   = (const float*)d_in[5];
    const float* yc    = (const float*)d_in[6];
    sa_march_kernel<<<dim3(NBATCH), dim3(BLK), 0, stream>>>(
        nuHat, u0t, v0t, dxl, dyv, dyd, yc, (float*)d_out);
}